// LayoutGraphModel_64132451664402
// MI455X (gfx1250) — compile-verified
//
#include <hip/hip_runtime.h>
#include <stdint.h>

typedef float v2f __attribute__((ext_vector_type(2)));
typedef float v8f __attribute__((ext_vector_type(8)));
typedef unsigned int u32x4 __attribute__((ext_vector_type(4)));
typedef int i32x4 __attribute__((ext_vector_type(4)));
typedef int i32x8 __attribute__((ext_vector_type(8)));
typedef __attribute__((address_space(3))) void lds_void;

#define WMMA_F32_4(A, B, C) \
  __builtin_amdgcn_wmma_f32_16x16x4_f32(false, (A), false, (B), (short)0, (C), false, false)

// ---------------------------------------------------------------------------
// Node features: x0[:,0:64] = emb_table[label], x0[:,64:128] = MLP(bbox)
// One wave handles 16 nodes. bbox@W1 is a native 16x16x4 f32 WMMA (K=4).
// ---------------------------------------------------------------------------
__global__ __launch_bounds__(256)
void feat_kernel(const int* __restrict__ label, const float* __restrict__ bbox,
                 const float* __restrict__ emb,
                 const float* __restrict__ w1, const float* __restrict__ b1,
                 const float* __restrict__ w2, const float* __restrict__ b2,
                 float* __restrict__ x0, int n) {
  __shared__ float hs[8][16][68];          // per-wave 16x64 h1 tile, padded
  const int wave = threadIdx.x >> 5;
  const int lane = threadIdx.x & 31;
  const int m    = lane & 15;
  const int kh   = lane >> 4;              // 0: K low half, 1: K high half
  const int base = (blockIdx.x * 8 + wave) * 16;

  // Embedding gather -> x0[:, 0:64] (float4, coalesced per row)
  for (int i = lane; i < 256; i += 32) {
    int r = i >> 4, c4 = i & 15;
    int g = base + r;
    if (g < n) {
      int lb = label[g];
      float4 v = *(const float4*)(emb + (size_t)lb * 64 + c4 * 4);
      *(float4*)(x0 + (size_t)g * 128 + c4 * 4) = v;
    }
  }

  // Layer 1: h1 = relu(bbox(16x4) @ w1(4x64) + b1) via 4 WMMAs (K=4 exact)
  v2f a = {};
  {
    int gm = base + m;
    if (gm < n) {
      a.x = bbox[gm * 4 + 2 * kh];
      a.y = bbox[gm * 4 + 2 * kh + 1];
    }
  }
#pragma unroll
  for (int nt = 0; nt < 4; ++nt) {
    int col = nt * 16 + m;
    v2f b;
    b.x = w1[(2 * kh) * 64 + col];
    b.y = w1[(2 * kh + 1) * 64 + col];
    v8f c = {};
    c = WMMA_F32_4(a, b, c);
    float bias = b1[col];
#pragma unroll
    for (int r = 0; r < 8; ++r)
      hs[wave][r + 8 * kh][col] = fmaxf(c[r] + bias, 0.0f);
  }
  __syncthreads();

  // Layer 2: h2 = h1(16x64) @ w2(64x64) + b2 -> x0[:,64:128]
  v8f acc[4] = {};
#pragma unroll
  for (int ks = 0; ks < 16; ++ks) {
    int k0 = ks * 4 + kh * 2;
    v2f a2;
    a2.x = hs[wave][m][k0];
    a2.y = hs[wave][m][k0 + 1];
#pragma unroll
    for (int nt = 0; nt < 4; ++nt) {
      int col = nt * 16 + m;
      v2f b;
      b.x = w2[k0 * 64 + col];
      b.y = w2[(k0 + 1) * 64 + col];
      acc[nt] = WMMA_F32_4(a2, b, acc[nt]);
    }
  }
#pragma unroll
  for (int nt = 0; nt < 4; ++nt) {
    int col = nt * 16 + m;
    float bias = b2[col];
#pragma unroll
    for (int r = 0; r < 8; ++r) {
      int g = base + r + 8 * kh;
      if (g < n) x0[(size_t)g * 128 + 64 + col] = acc[nt][r] + bias;
    }
  }
}

// ---------------------------------------------------------------------------
// GCN GEMM: h = x @ W, x [N,128], W [128,128]. 4 waves/block, 64-row tile.
// x tile is DMA'd into LDS by the Tensor Data Mover with 4-DWORD row padding
// (row stride 132 -> conflict-free ds_load_b64 A-fragments). B from global (L1).
// ---------------------------------------------------------------------------
__global__ __launch_bounds__(128)
void gcn_gemm_kernel(const float* __restrict__ x, const float* __restrict__ W,
                     float* __restrict__ h, int n) {
  __shared__ float xs[64][132];
  const int tid  = threadIdx.x;
  const int row0 = blockIdx.x * 64;

  if (tid < 32) {   // wave 0 issues the TDM descriptor
    unsigned lds_off = (unsigned)(uintptr_t)(lds_void*)&xs[0][0];
    unsigned long long ga =
        (unsigned long long)(uintptr_t)(x + (size_t)row0 * 128);
    unsigned rows_rem = (unsigned)(n - row0);   // grid guarantees >= 1

    u32x4 g0;
    g0[0] = 1u;                                  // count=1, user descriptor
    g0[1] = lds_off;                             // lds_addr (bytes)
    g0[2] = (unsigned)ga;                        // global_addr[31:0]
    g0[3] = (unsigned)(ga >> 32) | (2u << 30);   // global_addr[56:32] | type=2

    i32x8 g1;
    // data_size=4B (2<<16) | pad_enable (1<<20) | pad_interval=6 (128 DWORDs)
    // | pad_amount=3 (4 DWORDs) -> LDS row stride 132 DWORDs
    g1[0] = (int)((2u << 16) | (1u << 20) | (6u << 22) | (3u << 25));
    g1[1] = (int)(128u << 16);                   // tensor_dim0[15:0] = 128
    g1[2] = (int)((rows_rem & 0xffffu) << 16);   // tensor_dim1[15:0]
    g1[3] = (int)((rows_rem >> 16) | (128u << 16)); // tensor_dim1[31:16] | tile_dim0=128
    g1[4] = 64;                                  // tile_dim1 = 64 rows
    g1[5] = 128;                                 // tensor_dim0_stride = 128 elems
    g1[6] = 0;
    g1[7] = 0;

    i32x4 z4 = {0, 0, 0, 0};                     // groups 2/3 unused (2D tensor)
    i32x8 z8 = {0, 0, 0, 0, 0, 0, 0, 0};
    __builtin_amdgcn_tensor_load_to_lds(g0, g1, z4, z4, z8, 0);
    __builtin_amdgcn_s_wait_tensorcnt(0);
  }
  __syncthreads();

  const int wave = tid >> 5;
  const int lane = tid & 31;
  const int m    = lane & 15;
  const int kh   = lane >> 4;
  const int wr   = wave * 16;

  v8f acc[8] = {};
#pragma unroll
  for (int ks = 0; ks < 32; ++ks) {
    int k0 = ks * 4 + kh * 2;
    v2f a;
    a.x = xs[wr + m][k0];
    a.y = xs[wr + m][k0 + 1];
#pragma unroll
    for (int nt = 0; nt < 8; ++nt) {
      int col = nt * 16 + m;
      v2f b;
      b.x = W[k0 * 128 + col];
      b.y = W[(k0 + 1) * 128 + col];
      acc[nt] = WMMA_F32_4(a, b, acc[nt]);
    }
  }
#pragma unroll
  for (int nt = 0; nt < 8; ++nt) {
#pragma unroll
    for (int r = 0; r < 8; ++r) {
      int g = row0 + wr + r + 8 * kh;
      if (g < n) h[(size_t)g * 128 + nt * 16 + m] = acc[nt][r];
    }
  }
}

// ---------------------------------------------------------------------------
// Degree / normalization
// ---------------------------------------------------------------------------
__global__ void deg_init_kernel(float* __restrict__ deg, int n) {
  int i = blockIdx.x * blockDim.x + threadIdx.x;
  if (i < n) deg[i] = 1.0f;                    // self loop
}
__global__ void deg_accum_kernel(const int* __restrict__ ei, float* __restrict__ deg, int e) {
  int t = blockIdx.x * blockDim.x + threadIdx.x;
  if (t < e) atomicAdd(&deg[ei[e + t]], 1.0f); // dst row of edge_index
}
__global__ void dinv_kernel(float* __restrict__ deg, int n) {
  int i = blockIdx.x * blockDim.x + threadIdx.x;
  if (i < n) {
    float d = deg[i];
    deg[i] = d > 0.0f ? rsqrtf(d) : 0.0f;
  }
}

// ---------------------------------------------------------------------------
// Aggregation: out = dinv[i]^2 * h[i] (self loop, also init), then edge scatter
// ---------------------------------------------------------------------------
__global__ void self_loop_kernel(const float* __restrict__ h, const float* __restrict__ dinv,
                                 float* __restrict__ out, int n) {
  int t = blockIdx.x * blockDim.x + threadIdx.x;
  int i = t >> 5, lane = t & 31;
  if (i >= n) return;
  float w = dinv[i] * dinv[i];
  float4 v = *(const float4*)(h + (size_t)i * 128 + lane * 4);
  float4 o = {v.x * w, v.y * w, v.z * w, v.w * w};
  *(float4*)(out + (size_t)i * 128 + lane * 4) = o;
}

__global__ void edge_scatter_kernel(const int* __restrict__ ei, const float* __restrict__ h,
                                    const float* __restrict__ dinv, float* __restrict__ out,
                                    int e) {
  int t = blockIdx.x * blockDim.x + threadIdx.x;
  int edge = t >> 5, lane = t & 31;          // one wave per edge, 4 floats/lane
  if (edge >= e) return;
  int src = ei[edge];
  int dst = ei[e + edge];
  float w = dinv[src] * dinv[dst];
  float4 v = *(const float4*)(h + (size_t)src * 128 + lane * 4);
  float* o = out + (size_t)dst * 128 + lane * 4;
  atomicAdd(o + 0, v.x * w);
  atomicAdd(o + 1, v.y * w);
  atomicAdd(o + 2, v.z * w);
  atomicAdd(o + 3, v.w * w);
}

__global__ void bias_act_kernel(float* __restrict__ out, const float* __restrict__ b,
                                int n, int relu) {
  int t = blockIdx.x * blockDim.x + threadIdx.x;
  int i = t >> 5, lane = t & 31;
  if (i >= n) return;
  float4 v  = *(float4*)(out + (size_t)i * 128 + lane * 4);
  float4 bb = *(const float4*)(b + lane * 4);
  v.x += bb.x; v.y += bb.y; v.z += bb.z; v.w += bb.w;
  if (relu) {
    v.x = fmaxf(v.x, 0.f); v.y = fmaxf(v.y, 0.f);
    v.z = fmaxf(v.z, 0.f); v.w = fmaxf(v.w, 0.f);
  }
  *(float4*)(out + (size_t)i * 128 + lane * 4) = v;
}

// ---------------------------------------------------------------------------
extern "C" void kernel_launch(void* const* d_in, const int* in_sizes, int n_in,
                              void* d_out, int out_size, void* d_ws, size_t ws_size,
                              hipStream_t stream) {
  (void)n_in; (void)out_size; (void)ws_size;
  const int*   label = (const int*)d_in[0];
  const float* bbox  = (const float*)d_in[1];
  const int*   ei    = (const int*)d_in[2];
  const float* emb   = (const float*)d_in[3];
  const float* w1    = (const float*)d_in[4];
  const float* b1    = (const float*)d_in[5];
  const float* w2    = (const float*)d_in[6];
  const float* b2    = (const float*)d_in[7];
  const float* wg0   = (const float*)d_in[8];
  const float* bg0   = (const float*)d_in[9];
  const float* wg1   = (const float*)d_in[10];
  const float* bg1   = (const float*)d_in[11];
  const float* wg2   = (const float*)d_in[12];
  const float* bg2   = (const float*)d_in[13];
  const int n = in_sizes[0];
  const int e = in_sizes[2] / 2;
  float* out = (float*)d_out;

  float* x0  = (float*)d_ws;                 // [n,128]
  float* hb  = x0 + (size_t)n * 128;         // [n,128]
  float* ob  = hb + (size_t)n * 128;         // [n,128]
  float* deg = ob + (size_t)n * 128;         // [n]

  const dim3 b256(256), b128(128);
  const int featBlocks = (n + 127) / 128;
  const int nBlocks    = (n + 255) / 256;
  const int eBlocks    = (e + 255) / 256;
  const int gemmBlocks = (n + 63) / 64;
  const int rowBlocks  = (n * 32 + 255) / 256;
  const int edgeBlocks = (int)(((long long)e * 32 + 255) / 256);

  feat_kernel<<<featBlocks, b256, 0, stream>>>(label, bbox, emb, w1, b1, w2, b2, x0, n);
  deg_init_kernel<<<nBlocks, b256, 0, stream>>>(deg, n);
  deg_accum_kernel<<<eBlocks, b256, 0, stream>>>(ei, deg, e);
  dinv_kernel<<<nBlocks, b256, 0, stream>>>(deg, n);

  // Layer 0: x0 -> ob (relu)
  gcn_gemm_kernel<<<gemmBlocks, b128, 0, stream>>>(x0, wg0, hb, n);
  self_loop_kernel<<<rowBlocks, b256, 0, stream>>>(hb, deg, ob, n);
  edge_scatter_kernel<<<edgeBlocks, b256, 0, stream>>>(ei, hb, deg, ob, e);
  bias_act_kernel<<<rowBlocks, b256, 0, stream>>>(ob, bg0, n, 1);
  // Layer 1: ob -> x0 (relu, buffer reuse)
  gcn_gemm_kernel<<<gemmBlocks, b128, 0, stream>>>(ob, wg1, hb, n);
  self_loop_kernel<<<rowBlocks, b256, 0, stream>>>(hb, deg, x0, n);
  edge_scatter_kernel<<<edgeBlocks, b256, 0, stream>>>(ei, hb, deg, x0, e);
  bias_act_kernel<<<rowBlocks, b256, 0, stream>>>(x0, bg1, n, 1);
  // Layer 2: x0 -> d_out (no relu)
  gcn_gemm_kernel<<<gemmBlocks, b128, 0, stream>>>(x0, wg2, hb, n);
  self_loop_kernel<<<rowBlocks, b256, 0, stream>>>(hb, deg, out, n);
  edge_scatter_kernel<<<edgeBlocks, b256, 0, stream>>>(ei, hb, deg, out, e);
  bias_act_kernel<<<rowBlocks, b256, 0, stream>>>(out, bg2, n, 0);
}